// MaskBCELoss_19928648254234
// MI455X (gfx1250) — compile-verified
//
#include <hip/hip_runtime.h>

typedef __attribute__((ext_vector_type(2))) float v2f;
typedef __attribute__((ext_vector_type(8))) float v8f;

#define NUM_C 64
#define LOG2E 1.4426950408889634f
#define LN2   0.6931471805599453f

// Workspace float layout
#define WS_CNT 0      // 64: per-cluster counts
#define WS_SX  64     // 64: coord-x sums
#define WS_SY  128
#define WS_SZ  192
#define WS_SM  256    // 64: margin sums
#define WS_CX  320    // 64: centroid x
#define WS_CY  384
#define WS_CZ  448
#define WS_CSQ 512    // 64: |centroid|^2
#define WS_SIG 576    // 64: sigma
#define WS_INV 640    // 64: log2(e)/(2 sigma^2)
#define WS_ACC 704    // 3: [bce_sum, seed_sum, smooth_sum]
#define WS_FLOATS 707

// log(1 + exp(-p)) for p in [0,1]; even Taylor of ln(2 cosh(p/2)) - p/2.
// Max abs error ~2e-6 on [0,1].
__device__ __forceinline__ float softplus_neg01(float p) {
    float y = p * p;
    float r = __builtin_fmaf(-2.6351792e-5f, y, 3.4722222e-4f); // -17/645120, 1/2880
    r = __builtin_fmaf(r, y, -5.2083335e-3f);                   // -1/192
    r = __builtin_fmaf(r, y, 0.125f);                           //  1/8
    r = __builtin_fmaf(r, y, __builtin_fmaf(-0.5f, p, LN2));    //  ln2 - p/2
    return r;
}

__global__ void k_zero(float* ws) {
    for (int i = threadIdx.x; i < WS_FLOATS; i += blockDim.x) ws[i] = 0.0f;
}

// Segment sums: counts, coord sums, margin sums. LDS atomics then global flush.
__global__ void k_segsum(const float* __restrict__ coords,
                         const float* __restrict__ margins,
                         const int*   __restrict__ labels,
                         int n, float* __restrict__ ws) {
    __shared__ float s[5 * NUM_C]; // cnt | sx | sy | sz | sm
    for (int i = threadIdx.x; i < 5 * NUM_C; i += blockDim.x) s[i] = 0.0f;
    __syncthreads();
    for (int i = blockIdx.x * blockDim.x + threadIdx.x; i < n;
         i += gridDim.x * blockDim.x) {
        int lab = labels[i];
        atomicAdd(&s[lab],             1.0f);
        atomicAdd(&s[NUM_C     + lab], coords[3 * i + 0]);
        atomicAdd(&s[2 * NUM_C + lab], coords[3 * i + 1]);
        atomicAdd(&s[3 * NUM_C + lab], coords[3 * i + 2]);
        atomicAdd(&s[4 * NUM_C + lab], margins[i]);
    }
    __syncthreads();
    for (int i = threadIdx.x; i < 5 * NUM_C; i += blockDim.x)
        atomicAdd(&ws[i], s[i]);
}

// Per-cluster params: centroid, sigma, csq, log2e/(2 sigma^2)
__global__ void k_params(float* __restrict__ ws) {
    int c = threadIdx.x;
    if (c >= NUM_C) return;
    float cnt = ws[WS_CNT + c];
    float inv = 1.0f / cnt;
    float cx = ws[WS_SX + c] * inv;
    float cy = ws[WS_SY + c] * inv;
    float cz = ws[WS_SZ + c] * inv;
    float sg = ws[WS_SM + c] * inv;
    ws[WS_CX  + c] = cx;
    ws[WS_CY  + c] = cy;
    ws[WS_CZ  + c] = cz;
    ws[WS_CSQ + c] = cx * cx + cy * cy + cz * cz;
    ws[WS_SIG + c] = sg;
    ws[WS_INV + c] = LOG2E / (2.0f * sg * sg);
}

// Main loss kernel. One wave handles 16 points x all 64 clusters per tile via
// 4x V_WMMA_F32_16X16X4_F32. A = [cx cy cz -csq/2] (16 clusters x K4),
// B = [e0 e1 e2 1]^T (K4 x 16 points) => D = c.e - csq/2.
__global__ void __launch_bounds__(256)
k_main(const float* __restrict__ emb,
       const float* __restrict__ margins,
       const float* __restrict__ seed,
       const int*   __restrict__ labels,
       int n, float* __restrict__ ws) {
    __shared__ float s_cx[NUM_C], s_cy[NUM_C], s_cz[NUM_C];
    __shared__ float s_csq[NUM_C], s_inv2[NUM_C], s_sig[NUM_C];
    __shared__ float red[256];

    if (threadIdx.x < NUM_C) {
        s_cx[threadIdx.x]   = ws[WS_CX  + threadIdx.x];
        s_cy[threadIdx.x]   = ws[WS_CY  + threadIdx.x];
        s_cz[threadIdx.x]   = ws[WS_CZ  + threadIdx.x];
        s_csq[threadIdx.x]  = ws[WS_CSQ + threadIdx.x];
        s_inv2[threadIdx.x] = ws[WS_INV + threadIdx.x];
        s_sig[threadIdx.x]  = ws[WS_SIG + threadIdx.x];
    }
    __syncthreads();

    const int lane = threadIdx.x & 31;
    const bool hi  = lane >= 16;
    const int  lp  = lane & 15;

    // A-matrix (16x4 f32): lanes 0-15 hold {K0,K1} = {cx,cy},
    // lanes 16-31 hold {K2,K3} = {cz, -csq/2}. Row M = lp in tile t.
    v2f a[4];
#pragma unroll
    for (int t = 0; t < 4; ++t) {
        int m = 16 * t + lp;
        a[t].x = hi ? s_cz[m] : s_cx[m];
        a[t].y = hi ? (-0.5f * s_csq[m]) : s_cy[m];
    }

    // Hoist loop-invariant per-lane sigma scales into registers.
    float inv2r[32];
#pragma unroll
    for (int t = 0; t < 4; ++t)
#pragma unroll
        for (int r = 0; r < 8; ++r)
            inv2r[t * 8 + r] = s_inv2[16 * t + r + (hi ? 8 : 0)];

    const int waveId = (blockIdx.x * blockDim.x + threadIdx.x) >> 5;
    const int nWaves = (gridDim.x * blockDim.x) >> 5;
    const int tiles  = n >> 4;

    float bce = 0.0f, sdl = 0.0f, sml = 0.0f;

    for (int tile = waveId; tile < tiles; tile += nWaves) {
        int pt = tile * 16 + lp;
        float e0 = emb[3 * pt + 0];
        float e1 = emb[3 * pt + 1];
        float e2 = emb[3 * pt + 2];
        // B-matrix (4x16 f32): VGPR0 = {K0=e0 lo | K2=e2 hi},
        //                      VGPR1 = {K1=e1 lo | K3=1.0 hi}
        v2f b;
        b.x = hi ? e2 : e0;
        b.y = hi ? 1.0f : e1;
        float esq = e0 * e0 + e1 * e1 + e2 * e2;

        v8f czero = {};
#pragma unroll
        for (int t = 0; t < 4; ++t) {
            v8f d = __builtin_amdgcn_wmma_f32_16x16x4_f32(
                false, a[t], false, b, (short)0, czero, false, false);
#pragma unroll
            for (int r = 0; r < 8; ++r) {
                // dist = csq + esq - 2 c.e ; d already = c.e - csq/2
                float t0  = __builtin_fmaf(2.0f, d[r], -esq); // = -dist
                float arg = t0 * inv2r[t * 8 + r];            // log2-domain
                float p   = fminf(__builtin_amdgcn_exp2f(arg), 1.0f);
                bce += p + softplus_neg01(p);                 // label term handled below
            }
        }
        if (!hi) { // once per point: label-prob correction, seediness, smoothing
            int lab   = labels[pt];
            float sd  = seed[pt];
            float mg  = margins[pt];
            float dot = __builtin_fmaf(e0, s_cx[lab],
                        __builtin_fmaf(e1, s_cy[lab], e2 * s_cz[lab]));
            float argl = (__builtin_fmaf(2.0f, dot, -s_csq[lab]) - esq) * s_inv2[lab];
            float pl   = fminf(__builtin_amdgcn_exp2f(argl), 1.0f);
            bce -= pl;                 // subtract p*mask contribution
            sdl += fabsf(pl - sd);
            float dd = mg - s_sig[lab];
            sml += dd * dd;
        }
    }

    // Scalar tail for n % 16 points (dead for N=1M), outside the WMMA loop.
    int rem = n & 15;
    if (blockIdx.x == 0 && (int)threadIdx.x < rem) {
        int pt = (n & ~15) + threadIdx.x;
        float e0 = emb[3 * pt + 0];
        float e1 = emb[3 * pt + 1];
        float e2 = emb[3 * pt + 2];
        float esq = e0 * e0 + e1 * e1 + e2 * e2;
        int lab   = labels[pt];
        float sd  = seed[pt];
        for (int m = 0; m < NUM_C; ++m) {
            float dot = __builtin_fmaf(e0, s_cx[m],
                        __builtin_fmaf(e1, s_cy[m], e2 * s_cz[m]));
            float arg = (__builtin_fmaf(2.0f, dot, -s_csq[m]) - esq) * s_inv2[m];
            float p   = fminf(__builtin_amdgcn_exp2f(arg), 1.0f);
            float match = (lab == m) ? 1.0f : 0.0f;
            bce += p - p * match + softplus_neg01(p);
            sdl += match * fabsf(p - sd);
        }
        float mg = margins[pt];
        float dd = mg - s_sig[lab];
        sml += dd * dd;
    }

    // Block reduction of the three partial sums, then global atomic add.
    float vals[3] = {bce, sdl, sml};
#pragma unroll
    for (int k = 0; k < 3; ++k) {
        __syncthreads();
        red[threadIdx.x] = vals[k];
        __syncthreads();
        for (int s = blockDim.x >> 1; s > 0; s >>= 1) {
            if ((int)threadIdx.x < s) red[threadIdx.x] += red[threadIdx.x + s];
            __syncthreads();
        }
        if (threadIdx.x == 0) atomicAdd(&ws[WS_ACC + k], red[0]);
    }
}

__global__ void k_final(const float* __restrict__ ws, float* __restrict__ out, int n) {
    float mask_loss   = ws[WS_ACC + 0] / ((float)NUM_C * (float)n);
    float seed_loss   = ws[WS_ACC + 1] / (float)n;
    float smooth_loss = ws[WS_ACC + 2] / (float)NUM_C;
    out[0] = mask_loss + seed_loss + smooth_loss;
}

extern "C" void kernel_launch(void* const* d_in, const int* in_sizes, int n_in,
                              void* d_out, int out_size, void* d_ws, size_t ws_size,
                              hipStream_t stream) {
    const float* emb     = (const float*)d_in[0]; // (N,3)
    const float* margins = (const float*)d_in[1]; // (N,1)
    const float* seed    = (const float*)d_in[2]; // (N,1)
    const float* coords  = (const float*)d_in[3]; // (N,3)
    const int*   labels  = (const int*)d_in[4];   // (N,)
    float* out = (float*)d_out;
    float* ws  = (float*)d_ws;
    int n = in_sizes[4];

    k_zero<<<1, 256, 0, stream>>>(ws);
    k_segsum<<<1024, 256, 0, stream>>>(coords, margins, labels, n, ws);
    k_params<<<1, 64, 0, stream>>>(ws);
    k_main<<<1024, 256, 0, stream>>>(emb, margins, seed, labels, n, ws);
    k_final<<<1, 1, 0, stream>>>(ws, out, n);
}